// SSIM_3D_77867757077171
// MI455X (gfx1250) — compile-verified
//
#include <hip/hip_runtime.h>

typedef float v2f __attribute__((ext_vector_type(2)));
typedef float v8f __attribute__((ext_vector_type(8)));

// Output tile per block
#define TD 4
#define TH 16
#define TW 16
#define ID_ (TD + 10)      // 14 input d-planes
#define IH_ (TH + 10)      // 26 input h-rows
#define ROWS (ID_ * IH_)   // 364 (d,h) rows for the W-pass
#define PA 17              // LDS pitch for sA (odd -> fewer bank conflicts)
#define KTOT 28            // padded K for 7 chained 16x16x4 WMMAs

// Problem dims (from reference): img (2,1,96,256,256), window 11^3, VALID conv
#define DIM_D 96
#define DIM_H 256
#define DIM_W 256
#define OD 86
#define OH 246
#define OW 246
#define TOTAL_OUT 10408752.0f   // 2*86*246*246

__global__ __launch_bounds__(128) void ssim3d_main(
    const float* __restrict__ img1, const float* __restrict__ img2,
    const float* __restrict__ win, float* __restrict__ partials,
    float* __restrict__ out) {
  __shared__ float g1d[16];            // separable 1D gaussian (marginal of 3D win)
  __shared__ float bg[KTOT * 16];      // banded matrix  bg[K*16+w] = g[K-w]
  __shared__ float sA[ROWS * PA];      // after W-conv: rows=(d,h), cols=w
  __shared__ float sB[ID_ * TH * TW];  // after H-conv
  __shared__ float res[5 * TD * TH * TW];
  __shared__ float red[128];

  const int tid = threadIdx.x;
  const int wid = tid >> 5;
  const int lane = tid & 31;
  const int m = lane & 15;                 // matrix row/col index for this lane
  const int koff = (lane < 16) ? 0 : 2;    // K sub-slot per lane half (ISA layout)
  const int rowadd = (lane < 16) ? 0 : 8;  // C/D row offset per lane half

  const int bw = blockIdx.x;   // 16 tiles of W
  const int bh = blockIdx.y;   // 16 tiles of H
  const int bzn = blockIdx.z;  // 22 d-tiles * 2 batches
  const int bd = bzn % 22;
  const int n = bzn / 22;

  const int w0 = bw * TW;
  const int h0 = bh * TH;
  const int d0 = bd * TD;

  // --- recover the 1D kernel: the 3D window is separable; marginal over (d,h)
  if (tid < 11) {
    float s = 0.f;
    for (int ab = 0; ab < 121; ++ab) s += win[ab * 11 + tid];
    g1d[tid] = s;
  }
  __syncthreads();
  for (int i = tid; i < KTOT * 16; i += 128) {
    int K = i >> 4, w = i & 15, t = K - w;
    bg[i] = (t >= 0 && t <= 10) ? g1d[t] : 0.f;
  }
  __syncthreads();

  // Preload the banded-g WMMA operand once: lane expression bg[K*16+m] is both
  // the W-pass B-matrix (B[K][n], n=m) and the H-pass A-matrix (A[h][K], h=m).
  v2f bgv[7];
#pragma unroll
  for (int kk = 0; kk < 7; ++kk) {
    int K0 = kk * 4 + koff;
    bgv[kk].x = bg[K0 * 16 + m];
    bgv[kk].y = bg[(K0 + 1) * 16 + m];
  }

  const size_t base_n = (size_t)n * DIM_D * DIM_H * DIM_W;
  const int wmax = (DIM_W - 1) - w0;             // clamp limit along W
  const bool fastW = (w0 + KTOT <= DIM_W);       // interior W-tile: no clamping

  for (int f = 0; f < 5; ++f) {
    const bool need1 = (f == 0) || (f == 2) || (f == 4);
    const bool need2 = (f == 1) || (f == 3) || (f == 4);

    // -------- W-pass: out(16 rows x 16 w) = field(16x28) * bg(28x16) ----------
    for (int rt = wid; rt * 16 < ROWS; rt += 4) {
      int r = rt * 16 + m;
      int rcl = (r < ROWS) ? r : (ROWS - 1);
      int rd = rcl / IH_;
      int rh = rcl - rd * IH_;
      int id = d0 + rd;  if (id > DIM_D - 1) id = DIM_D - 1;
      int ih = h0 + rh;  if (ih > DIM_H - 1) ih = DIM_H - 1;
      const float* p1 =
          img1 + (base_n + ((size_t)id << 16) + ((size_t)ih << 8) + (size_t)w0);
      const float* p2 =
          img2 + (base_n + ((size_t)id << 16) + ((size_t)ih << 8) + (size_t)w0);

      // 1) Load raw samples for this lane's 14 K-slots (clauses up, one wait).
      float x1[14], x2[14];
      if (fastW) {
#pragma unroll
        for (int kk = 0; kk < 7; ++kk) {
          int o = kk * 4 + koff;  // even -> 8B aligned
          if (need1) {
            float2 t = *(const float2*)(p1 + o);
            x1[2 * kk] = t.x; x1[2 * kk + 1] = t.y;
          }
          if (need2) {
            float2 t = *(const float2*)(p2 + o);
            x2[2 * kk] = t.x; x2[2 * kk + 1] = t.y;
          }
        }
      } else {
#pragma unroll
        for (int kk = 0; kk < 7; ++kk) {
          int K0 = kk * 4 + koff, K1 = K0 + 1;
          int o0 = (K0 < wmax) ? K0 : wmax;
          int o1 = (K1 < wmax) ? K1 : wmax;
          if (need1) { x1[2 * kk] = p1[o0]; x1[2 * kk + 1] = p1[o1]; }
          if (need2) { x2[2 * kk] = p2[o0]; x2[2 * kk + 1] = p2[o1]; }
        }
      }

      // 2) Branchless field combine (selectors are wave-uniform).
      float av[14];
#pragma unroll
      for (int i = 0; i < 14; ++i) {
        float a0 = need1 ? x1[i] : x2[i];
        float b0 = (f < 2) ? 1.f : ((f == 2) ? x1[i] : x2[i]);
        av[i] = a0 * b0;
      }

      // 3) Pure-register 7-chain WMMA.
      v8f c = {0.f, 0.f, 0.f, 0.f, 0.f, 0.f, 0.f, 0.f};
#pragma unroll
      for (int kk = 0; kk < 7; ++kk) {
        v2f a; a.x = av[2 * kk]; a.y = av[2 * kk + 1];
        c = __builtin_amdgcn_wmma_f32_16x16x4_f32(false, a, false, bgv[kk],
                                                  (short)0, c, false, false);
      }

      // 4) Store D-tile (guard only needed on the final partial row-tile).
      if (rt * 16 + 16 <= ROWS) {
#pragma unroll
        for (int i = 0; i < 8; ++i) sA[(rt * 16 + i + rowadd) * PA + m] = c[i];
      } else {
#pragma unroll
        for (int i = 0; i < 8; ++i) {
          int rr = rt * 16 + i + rowadd;
          if (rr < ROWS) sA[rr * PA + m] = c[i];
        }
      }
    }
    __syncthreads();

    // -------- H-pass: out(16h x 16w) = bg^T(16x28) * sA(28x16) ---------------
    for (int dp = wid; dp < ID_; dp += 4) {
      const int rowb = dp * IH_;
      v8f c = {0.f, 0.f, 0.f, 0.f, 0.f, 0.f, 0.f, 0.f};
#pragma unroll
      for (int kk = 0; kk < 7; ++kk) {
        int K0 = kk * 4 + koff;
        int kc0 = (K0 < IH_) ? K0 : (IH_ - 1);      // g coeff is 0 there anyway
        int kc1 = (K0 + 1 < IH_) ? (K0 + 1) : (IH_ - 1);
        v2f b;
        b.x = sA[(rowb + kc0) * PA + m];
        b.y = sA[(rowb + kc1) * PA + m];
        c = __builtin_amdgcn_wmma_f32_16x16x4_f32(false, bgv[kk], false, b,
                                                  (short)0, c, false, false);
      }
#pragma unroll
      for (int i = 0; i < 8; ++i)
        sB[dp * (TH * TW) + (i + rowadd) * TW + m] = c[i];
    }
    __syncthreads();

    // -------- D-pass: 11-tap VALU FMA (cheap) --------------------------------
    for (int idx = tid; idx < TD * TH * TW; idx += 128) {
      int dd = idx >> 8;
      int hw = idx & 255;
      float acc = 0.f;
#pragma unroll
      for (int t = 0; t < 11; ++t) acc += sB[(dd + t) * 256 + hw] * g1d[t];
      res[f * (TD * TH * TW) + idx] = acc;
    }
    __syncthreads();
  }

  // -------- SSIM combine + block reduction -----------------------------------
  float lsum = 0.f;
  for (int idx = tid; idx < TD * TH * TW; idx += 128) {
    int dd = idx >> 8, hw = idx & 255, hh = hw >> 4, ww = hw & 15;
    int od = d0 + dd, oh = h0 + hh, ow = w0 + ww;
    if (od < OD && oh < OH && ow < OW) {
      float mu1 = res[idx];
      float mu2 = res[1024 + idx];
      float x11 = res[2048 + idx];
      float x22 = res[3072 + idx];
      float x12 = res[4096 + idx];
      float mu1s = mu1 * mu1, mu2s = mu2 * mu2, m12 = mu1 * mu2;
      float s1 = x11 - mu1s, s2 = x22 - mu2s, s12 = x12 - m12;
      const float C1 = 1e-4f, C2 = 9e-4f;
      float numv = (2.f * m12 + C1) * (2.f * s12 + C2);
      float denv = (mu1s + mu2s + C1) * (s1 + s2 + C2);
      lsum += numv / denv;
    }
  }
  red[tid] = lsum;
  __syncthreads();
  for (int s = 64; s > 0; s >>= 1) {
    if (tid < s) red[tid] += red[tid + s];
    __syncthreads();
  }
  if (tid == 0) {
    int blkId = (blockIdx.z * 16 + blockIdx.y) * 16 + blockIdx.x;
    if (partials)
      partials[blkId] = red[0];
    else
      atomicAdd(out, red[0] * (1.f / TOTAL_OUT));
  }
}

__global__ void ssim3d_reduce(const float* __restrict__ partials, int nParts,
                              float* __restrict__ out) {
  __shared__ float red[256];
  float s = 0.f;
  for (int i = threadIdx.x; i < nParts; i += 256) s += partials[i];
  red[threadIdx.x] = s;
  __syncthreads();
  for (int t = 128; t > 0; t >>= 1) {
    if (threadIdx.x < t) red[threadIdx.x] += red[threadIdx.x + t];
    __syncthreads();
  }
  if (threadIdx.x == 0) out[0] = red[0] * (1.f / TOTAL_OUT);
}

__global__ void ssim3d_zero(float* out) { out[0] = 0.f; }

extern "C" void kernel_launch(void* const* d_in, const int* in_sizes, int n_in,
                              void* d_out, int out_size, void* d_ws,
                              size_t ws_size, hipStream_t stream) {
  const float* img1 = (const float*)d_in[0];
  const float* img2 = (const float*)d_in[1];
  const float* win = (const float*)d_in[2];
  float* out = (float*)d_out;

  dim3 grid(16, 16, 44);  // 16 W-tiles, 16 H-tiles, 22 D-tiles * 2 batches
  dim3 block(128);
  const int nParts = 16 * 16 * 44;

  bool useWs = ws_size >= (size_t)nParts * sizeof(float);
  float* parts = useWs ? (float*)d_ws : nullptr;
  if (!useWs) ssim3d_zero<<<1, 1, 0, stream>>>(out);
  ssim3d_main<<<grid, block, 0, stream>>>(img1, img2, win, parts, out);
  if (useWs) ssim3d_reduce<<<1, 256, 0, stream>>>(parts, nParts, out);
}